// ctc_loss_37589553774858
// MI455X (gfx1250) — compile-verified
//
#include <hip/hip_runtime.h>
#include <hip/hip_bf16.h>
#include <stdint.h>

namespace {

constexpr int kB = 32;
constexpr int kT = 1600;
constexpr int kV = 1024;
constexpr int kU = 128;
constexpr int kS = 2 * kU + 1;   // 257 extended states
constexpr float kNeg = -1e30f;

// Flat LDS address = {shared_aperture_hi32, lds_byte_offset}; the async DMA
// instructions want the per-lane LDS byte offset in a VGPR -> low 32 bits.
__device__ __forceinline__ unsigned lds_byte_off(const void* p) {
  return (unsigned)(unsigned long long)(uintptr_t)p;
}

// CDNA5 async global->LDS copy, 16 bytes per active lane. Tracked by ASYNCcnt.
__device__ __forceinline__ void async_row_b128(void* lds_dst, const void* gsrc) {
  asm volatile("global_load_async_to_lds_b128 %0, %1, off"
               :: "v"(lds_byte_off(lds_dst)), "v"(gsrc)
               : "memory");
}

__device__ __forceinline__ void wait_async_zero() {
  asm volatile("s_wait_asynccnt 0" ::: "memory");
}

__device__ __forceinline__ float logaddexp(float a, float b) {
  float m = fmaxf(a, b);
  float n = fminf(a, b);
  // n == m == kNeg  -> m + log(2)  (matches jnp.logaddexp semantics closely)
  return m + __logf(1.0f + __expf(n - m));
}

} // namespace

// One workgroup per batch element. 288 threads = 9 wave32 waves.
// Threads 0..256 own one alpha state; threads 0..255 drive the row DMA.
__global__ __launch_bounds__(288) void ctc_alpha_kernel(
    const float* __restrict__ outputs,      // [B,T,V] log-probs
    const int*   __restrict__ labels,       // [B,U]
    const int*   __restrict__ out_lens,     // [B]
    const int*   __restrict__ lab_lens,     // [B]
    float*       __restrict__ partial) {    // [B] per-batch -ll/len
  __shared__ float rowbuf[2][kV];           // double-buffered 4KB prob row
  __shared__ float alphaBuf[2][kS];         // ping-pong alpha
  __shared__ int   ext[kS];                 // extended label sequence

  const int b    = blockIdx.x;
  const int tid  = threadIdx.x;
  const int Tlen = out_lens[b];
  const int L    = lab_lens[b];
  const float* base = outputs + (size_t)b * kT * kV;

  // ext = [blank, l0, blank, l1, ..., blank]
  if (tid < kS)
    ext[tid] = (tid & 1) ? labels[(size_t)b * kU + (tid >> 1)] : 0;

  // DMA row 0 into LDS (256 lanes x 16B = 4KB)
  if (tid < kV / 4)
    async_row_b128(&rowbuf[0][tid * 4], base + tid * 4);
  wait_async_zero();
  __syncthreads();

  int  ext_s = 0;
  bool skip  = false;
  if (tid < kS) {
    ext_s = ext[tid];
    skip  = (tid >= 2) && (ext_s != 0) && (ext_s != ext[tid - 2]);
    float lp0 = rowbuf[0][ext_s];                 // LDS gather
    alphaBuf[0][tid] = (tid <= 1) ? lp0 : kNeg;   // only s=0,1 reachable at t=0
  }

  // prefetch row 1 while alpha0 settles
  if (tid < kV / 4)
    async_row_b128(&rowbuf[1][tid * 4], base + kV + tid * 4);

  int cur = 0;
  for (int t = 1; t < kT; ++t) {
    wait_async_zero();      // row t landed (and alpha writes fenced by barrier)
    __syncthreads();

    // software pipeline: kick off DMA of row t+1 into the other buffer
    if (t + 1 < kT && tid < kV / 4)
      async_row_b128(&rowbuf[(t + 1) & 1][tid * 4],
                     base + (size_t)(t + 1) * kV + tid * 4);

    if (tid < kS) {
      const float* A = alphaBuf[cur];
      float a1 = A[tid];
      float a2 = (tid >= 1) ? A[tid - 1] : kNeg;
      float a3 = skip ? A[tid - 2] : kNeg;
      float lp = rowbuf[t & 1][ext_s];            // LDS gather of log-prob
      float nv = logaddexp(logaddexp(a1, a2), a3) + lp;
      alphaBuf[cur ^ 1][tid] = (t < Tlen) ? nv : a1;  // freeze past seq length
    }
    cur ^= 1;
  }
  __syncthreads();

  if (tid == 0) {
    const float* A = alphaBuf[cur];
    int   last = 2 * L;
    float aN   = A[last];
    float aN1  = A[last > 0 ? last - 1 : 0];
    float ll   = logaddexp(aN, aN1);
    partial[b] = -ll / (float)L;
  }
}

__global__ void ctc_reduce_kernel(const float* __restrict__ partial,
                                  float* __restrict__ out) {
  if (threadIdx.x == 0) {
    float s = 0.f;
    for (int i = 0; i < kB; ++i) s += partial[i];
    out[0] = s / (float)kB;
  }
}

extern "C" void kernel_launch(void* const* d_in, const int* in_sizes, int n_in,
                              void* d_out, int out_size, void* d_ws, size_t ws_size,
                              hipStream_t stream) {
  const float* outputs  = (const float*)d_in[0];
  const int*   labels   = (const int*)d_in[1];
  const int*   out_lens = (const int*)d_in[2];
  const int*   lab_lens = (const int*)d_in[3];
  float* partial = (float*)d_ws;   // 32 floats of scratch

  ctc_alpha_kernel<<<kB, 288, 0, stream>>>(outputs, labels, out_lens, lab_lens,
                                           partial);
  ctc_reduce_kernel<<<1, 32, 0, stream>>>(partial, (float*)d_out);
}